// CartNN_83399674954403
// MI455X (gfx1250) — compile-verified
//
#include <hip/hip_runtime.h>
#include <math.h>

// out = tanh(tanh(x @ w + b)); x:[16384,4096] f32. Bandwidth-bound: 256 MB @ 23.3 TB/s ~= 11.5 us.
// Only 1024 waves exist (16 rows/wave via WMMA f32 16x16x4), so latency hiding must come from
// in-wave pipelining: TDM (tensor_load_to_lds) triple-buffers 16x128 f32 tiles into LDS
// (linear 512B bursts per row, ~17 KB in flight per wave), WMMA consumes from LDS via
// conflict-free padded ds_load_b128 (TDM pad: 4 DW per 128 DW row -> all 64 banks hit).

typedef __attribute__((ext_vector_type(2))) float v2f;
typedef __attribute__((ext_vector_type(4))) float v4f;
typedef __attribute__((ext_vector_type(8))) float v8f;
typedef __attribute__((ext_vector_type(4))) unsigned int v4u;
typedef __attribute__((ext_vector_type(4))) int v4i;
typedef __attribute__((ext_vector_type(8))) int v8i;

#define IN_SIZE        4096
#define BATCH_ROWS     16384
#define KC             128                    // K-chunk (floats) per TDM tile
#define NCHUNK         (IN_SIZE / KC)         // 32
#define ROWS_PER_WAVE  16
#define WAVES_PER_BLK  4
#define THREADS        (WAVES_PER_BLK * 32)   // 128
#define ROWS_PER_BLK   (ROWS_PER_WAVE * WAVES_PER_BLK) // 64
#define ROW_PITCH_B    (KC * 4 + 16)          // 528 B padded row pitch
#define TILE_B         (ROWS_PER_WAVE * ROW_PITCH_B)   // 8448 B
#define NBUF           3                      // triple buffer: 2 tiles in flight during compute

// Issue one TDM 2D tile load: 16 rows x 128 f32 from gsrc (row stride 4096 f32) into LDS.
// D# per CDNA5 ISA ch.8: group0 = {count/lds_addr/global_addr/type}, group1 = dims/strides/pad.
__device__ __forceinline__ void tdm_load_tile(const float* gsrc, unsigned lds_byte) {
  unsigned long long ga = (unsigned long long)(uintptr_t)gsrc;
  v4u g0;
  g0.x = 1u;                                            // count=1, no gather
  g0.y = lds_byte;                                      // lds_addr
  g0.z = (unsigned)ga;                                  // global_addr[31:0]
  g0.w = (unsigned)((ga >> 32) & 0x01FFFFFFu)           // global_addr[56:32]
       | (2u << 30);                                    // type = 2 ("image")
  v8i g1;
  g1[0] = (2 << 16)                                     // data_size = 4 B
        | (1 << 20)                                     // pad_enable
        | (6 << 22)                                     // pad_interval: every 128 DWORDs (512 B)
        | (3 << 25);                                    // pad_amount: 4 DWORDs (16 B)
  g1[1] = (int)((IN_SIZE & 0xFFFF) << 16);              // abar=0 | tensor_dim0[15:0]
  g1[2] = (int)((IN_SIZE >> 16) | ((BATCH_ROWS & 0xFFFF) << 16)); // dim0 hi | tensor_dim1 lo
  g1[3] = (int)((BATCH_ROWS >> 16) | (KC << 16));       // dim1 hi | tile_dim0 = 128
  g1[4] = ROWS_PER_WAVE;                                // tile_dim1 = 16, tile_dim2 = 0
  g1[5] = IN_SIZE;                                      // tensor_dim0_stride = 4096 (lo32)
  g1[6] = 0;                                            // stride hi | dim1_stride lo (2D: unused)
  g1[7] = 0;
  v4i gz4 = (v4i){0, 0, 0, 0};                          // groups 2/3 unused (<=2D tensor)
  v8i gz8 = (v8i){0, 0, 0, 0, 0, 0, 0, 0};
  __builtin_amdgcn_tensor_load_to_lds(g0, g1, gz4, gz4, gz8, 0);
}

__global__ __launch_bounds__(THREADS) void CartNN_gemv_tdm_wmma_tanh(
    const float* __restrict__ x,
    const float* __restrict__ w,
    const float* __restrict__ b,
    float* __restrict__ out)
{
  __shared__ float4 sW[IN_SIZE / 4];                                   // 16 KB
  __shared__ __align__(16) unsigned char sTile[WAVES_PER_BLK * NBUF * TILE_B]; // ~99 KB

  const int tid = threadIdx.x;

  // Stage w into LDS (coalesced float4), overlapped with priming the TDM pipe.
  const float4* w4 = reinterpret_cast<const float4*>(w);
  for (int i = tid; i < IN_SIZE / 4; i += THREADS) sW[i] = w4[i];

  const int wave  = __builtin_amdgcn_readfirstlane(tid >> 5); // wave-uniform for SGPR descriptors
  const int lane  = tid & 31;
  const int half  = lane >> 4;    // 0: lanes 0-15 (A K-slots 0,1) ; 1: lanes 16-31 (K-slots 2,3)
  const int rlane = lane & 15;

  const int row0 = (blockIdx.x * WAVES_PER_BLK + wave) * ROWS_PER_WAVE;
  const float* xrow = x + (size_t)row0 * IN_SIZE;
  const unsigned ldsBase = (unsigned)(uintptr_t)sTile + (unsigned)(wave * (NBUF * TILE_B));

  // Prime triple buffer: 3 tiles (25 KB) issued before first compute.
  tdm_load_tile(xrow + 0 * KC, ldsBase + 0 * TILE_B);
  tdm_load_tile(xrow + 1 * KC, ldsBase + 1 * TILE_B);
  tdm_load_tile(xrow + 2 * KC, ldsBase + 2 * TILE_B);

  __syncthreads();  // sW ready

  v8f acc = {};
  const unsigned laneOff = (unsigned)(rlane * ROW_PITCH_B + half * 16);

  int bi = 0;  // rotating buffer index 0..2 (uniform scalar)
  for (int c = 0; c < NCHUNK; ++c) {
    // TDM completes in order per wave: oldest outstanding tile is chunk c's.
    if (c < NCHUNK - 2)      __builtin_amdgcn_s_wait_tensorcnt(2);
    else if (c < NCHUNK - 1) __builtin_amdgcn_s_wait_tensorcnt(1);
    else                     __builtin_amdgcn_s_wait_tensorcnt(0);

    const unsigned char* tb = sTile + (unsigned)(wave * (NBUF * TILE_B))
                                    + (unsigned)(bi * TILE_B) + laneOff;
#pragma unroll 4
    for (int j = 0; j < KC / 8; ++j) {
      // One b128 LDS read supplies 8 K-values = 2 WMMAs (K-slot permutation mirrored in B).
      const v4f a = *reinterpret_cast<const v4f*>(tb + j * 32);
      const int w4i = (c * KC + j * 8) >> 2;
      const float4 wlo = sW[w4i];       // w[8j .. 8j+3]
      const float4 whi = sW[w4i + 1];   // w[8j+4 .. 8j+7]

      v2f aa1; aa1.x = a.x; aa1.y = a.y;
      v2f aa2; aa2.x = a.z; aa2.y = a.w;
      v2f bb1; bb1.x = half ? whi.x : wlo.x; bb1.y = half ? whi.y : wlo.y;
      v2f bb2; bb2.x = half ? whi.z : wlo.z; bb2.y = half ? whi.w : wlo.w;

      acc = __builtin_amdgcn_wmma_f32_16x16x4_f32(false, aa1, false, bb1,
                                                  (short)0, acc, false, false);
      acc = __builtin_amdgcn_wmma_f32_16x16x4_f32(false, aa2, false, bb2,
                                                  (short)0, acc, false, false);
    }

    // Refill the buffer just consumed (its ds reads were waited before the WMMAs that used them).
    if (c + NBUF < NCHUNK)
      tdm_load_tile(xrow + (c + NBUF) * KC, ldsBase + (unsigned)(bi * TILE_B));

    bi = (bi == NBUF - 1) ? 0 : bi + 1;
  }

  // D layout: lanes 0-15 VGPR r -> (M=r); lanes 16-31 -> (M=r+8); all N columns identical.
  const float bias = b[0];
  if (rlane == 0) {
    const int mbase = row0 + (half ? 8 : 0);
#pragma unroll
    for (int r = 0; r < 8; ++r) {
      const float h = tanhf(acc[r] + bias);
      out[mbase + r] = tanhf(h);
    }
  }
}

extern "C" void kernel_launch(void* const* d_in, const int* in_sizes, int n_in,
                              void* d_out, int out_size, void* d_ws, size_t ws_size,
                              hipStream_t stream) {
  const float* x = (const float*)d_in[0];
  const float* w = (const float*)d_in[1];
  const float* b = (const float*)d_in[2];
  float* out = (float*)d_out;

  const int batch  = in_sizes[0] / IN_SIZE;   // 16384
  const int blocks = batch / ROWS_PER_BLK;    // 256

  CartNN_gemv_tdm_wmma_tanh<<<blocks, THREADS, 0, stream>>>(x, w, b, out);
}